// GCNCritic_7980049236589
// MI455X (gfx1250) — compile-verified
//
#include <hip/hip_runtime.h>
#include <hip/hip_bf16.h>

typedef __attribute__((ext_vector_type(2))) float v2f;
typedef __attribute__((ext_vector_type(8))) float v8f;

// ---------------------------------------------------------------------------
// Kernel 1: fused node-mean reduction + fp32 WMMA GEMM over one K-chunk.
//   A[b,k] = (1/16) sum_i x[b,i,k]   (staged in LDS, 64 x 128 chunk)
//   partial[wg][b,h] = sum_{k in chunk} A[b,k] * W[k,h]
// 512 threads = 16 waves; each wave computes one 16x16 tile of the 64x64 out.
// ---------------------------------------------------------------------------
__global__ __launch_bounds__(512)
void k_gcn_fused(const float* __restrict__ x,     // [64,16,32768]
                 const float* __restrict__ W,     // [32768,64]
                 float* __restrict__ partials)    // [256,64,64]
{
    constexpr int KC  = 128;   // K-chunk per workgroup
    constexpr int KCP = 132;   // padded row stride (bank-conflict free, 16B aligned)
    __shared__ float Alds[64 * KCP];

    const int tid   = threadIdx.x;
    const int kbase = blockIdx.x * KC;

    // ---- Phase 1: reduce 16 nodes -> LDS (float4 streaming loads) ----------
    for (int o = tid; o < 64 * (KC / 4); o += 512) {
        const int b = o >> 5;        // KC/4 == 32 groups per batch row
        const int g = o & 31;
        const float* base = x + ((size_t)b * 16) * 32768 + (size_t)(kbase + g * 4);
        float s0 = 0.f, s1 = 0.f, s2 = 0.f, s3 = 0.f;
#pragma unroll
        for (int i = 0; i < 16; ++i) {
            const float4 v = *(const float4*)(base + (size_t)i * 32768);
            s0 += v.x; s1 += v.y; s2 += v.z; s3 += v.w;
        }
        float* dst = &Alds[b * KCP + g * 4];
        dst[0] = s0 * 0.0625f;
        dst[1] = s1 * 0.0625f;
        dst[2] = s2 * 0.0625f;
        dst[3] = s3 * 0.0625f;
    }
    __syncthreads();

    // ---- Phase 2: V_WMMA_F32_16X16X4_F32 over the K-chunk ------------------
    const int wave = tid >> 5;            // 0..15
    const int lane = tid & 31;
    const int mi   = wave >> 2;           // batch-tile 0..3
    const int ni   = wave & 3;            // hid-tile   0..3
    const int m    = lane & 15;           // M (and N-col for B) index
    const int kp   = (lane >> 4) << 1;    // K sub-offset: 0 (lanes 0-15) / 2 (16-31)

    const float* arow = &Alds[(mi * 16 + m) * KCP];
    const float* wcol = W + (size_t)kbase * 64 + (size_t)(ni * 16 + m);

    v8f c = {0.f, 0.f, 0.f, 0.f, 0.f, 0.f, 0.f, 0.f};
#pragma unroll 4
    for (int k0 = 0; k0 < KC; k0 += 4) {
        const int ka = k0 + kp;
        // A 16x4 layout: lane m holds K=ka (v0) and K=ka+1 (v1)
        v2f a;
        a.x = arow[ka];
        a.y = arow[ka + 1];
        // B 4x16 layout: VGPR j, lane -> K = ka + j, N = lane&15
        v2f bb;
        bb.x = wcol[(size_t)ka * 64];
        bb.y = wcol[(size_t)(ka + 1) * 64];
        c = __builtin_amdgcn_wmma_f32_16x16x4_f32(
                false, a, false, bb, (short)0, c, false, false);
    }

    // C/D layout: VGPR r -> M = r (+8 for lanes 16-31), N = lane&15
    float* pout = partials + (size_t)blockIdx.x * 4096 + (size_t)(ni * 16 + m);
    const int row0 = mi * 16 + ((lane >> 4) << 3);
#pragma unroll
    for (int r = 0; r < 8; ++r)
        pout[(size_t)(row0 + r) * 64] = c[r];
}

// ---------------------------------------------------------------------------
// Kernel 2: deterministic reduction of the 256 partial 64x64 tiles.
// ---------------------------------------------------------------------------
__global__ __launch_bounds__(256)
void k_reduce_partials(const float* __restrict__ partials,  // [256,4096]
                       float* __restrict__ pooled)          // [4096]
{
    const int j = blockIdx.x * 256 + threadIdx.x;  // 0..4095
    float s = 0.f;
#pragma unroll 8
    for (int w = 0; w < 256; ++w)
        s += partials[(size_t)w * 4096 + j];
    pooled[j] = s;
}

// ---------------------------------------------------------------------------
// Kernel 3: per-agent critic heads + argmax-gather. 1 block per agent,
// 1 thread per batch row. Tiny (256 threads total).
// ---------------------------------------------------------------------------
__global__ __launch_bounds__(64)
void k_heads(const float* __restrict__ pooled,   // [64,64] (pre-bias)
             const float* __restrict__ gcn_b,    // [64]
             const float* __restrict__ actions,  // [4,64,13]
             const float* __restrict__ w1,       // [4,64,64]
             const float* __restrict__ b1,       // [4,64]
             const float* __restrict__ w2,       // [4,64,13]
             const float* __restrict__ b2,       // [4,13]
             float* __restrict__ out)            // [4,64]
{
    const int a = blockIdx.x;
    const int b = threadIdx.x;

    float p[64];
#pragma unroll
    for (int h = 0; h < 64; ++h)
        p[h] = pooled[b * 64 + h] + gcn_b[h];

    float z[64];
    for (int ho = 0; ho < 64; ++ho) {
        float s = b1[a * 64 + ho];
#pragma unroll 8
        for (int hi = 0; hi < 64; ++hi)
            s += p[hi] * w1[((size_t)(a * 64 + hi)) * 64 + ho];
        z[ho] = (s >= 0.f) ? s : 0.01f * s;   // LeakyReLU(0.01)
    }

    // first-occurrence argmax over 13 actions (matches jnp.argmax)
    const float* arow = actions + (size_t)(a * 64 + b) * 13;
    int   jbest = 0;
    float vbest = arow[0];
#pragma unroll
    for (int jj = 1; jj < 13; ++jj) {
        const float v = arow[jj];
        if (v > vbest) { vbest = v; jbest = jj; }
    }

    float q = b2[a * 13 + jbest];
#pragma unroll 8
    for (int h = 0; h < 64; ++h)
        q += z[h] * w2[((size_t)(a * 64 + h)) * 13 + jbest];

    out[a * 64 + b] = q;
}

// ---------------------------------------------------------------------------
extern "C" void kernel_launch(void* const* d_in, const int* in_sizes, int n_in,
                              void* d_out, int out_size, void* d_ws, size_t ws_size,
                              hipStream_t stream)
{
    (void)in_sizes; (void)n_in; (void)out_size; (void)ws_size;

    const float* x       = (const float*)d_in[0];  // unary_tensor [64,16,32768]
    // d_in[1] = binary_tensor (unused by the reference)
    const float* actions = (const float*)d_in[2];  // [4,64,13]
    const float* gcn_w   = (const float*)d_in[3];  // [32768,64]
    const float* gcn_b   = (const float*)d_in[4];  // [64]
    const float* hw1     = (const float*)d_in[5];  // [4,64,64]
    const float* hb1     = (const float*)d_in[6];  // [4,64]
    const float* hw2     = (const float*)d_in[7];  // [4,64,13]
    const float* hb2     = (const float*)d_in[8];  // [4,13]

    float* partials = (float*)d_ws;                 // 256*4096 floats = 4 MiB
    float* pooled   = partials + 256 * 4096;        // 4096 floats

    k_gcn_fused     <<<256, 512, 0, stream>>>(x, gcn_w, partials);
    k_reduce_partials<<<16, 256, 0, stream>>>(partials, pooled);
    k_heads         <<<4,   64, 0, stream>>>(pooled, gcn_b, actions,
                                             hw1, hb1, hw2, hb2, (float*)d_out);
}